// GCN_73650099192275
// MI455X (gfx1250) — compile-verified
//
#include <hip/hip_runtime.h>
#include <math.h>

#define N_NODES   50000
#define N_EDGES   800000
#define N_GRAPHS  500
#define HID       64
#define KQVS      256      // k|q|v|skip concatenated
#define N_CLASSES 10

typedef __attribute__((ext_vector_type(2))) float v2f;
typedef __attribute__((ext_vector_type(8))) float v8f;

// ---------------------------------------------------------------- utilities
__global__ void zero_f32(float* __restrict__ p, int n) {
  int i = blockIdx.x * blockDim.x + threadIdx.x;
  if (i < n) p[i] = 0.0f;
}

// ------------------------------------------------- layer 1: IN_FEAT==1 path
// kqvs[n, c] = x[n] * Wcat[c] + bcat[c]   (Wcat = W1 flattened [4*64]=[256])
__global__ void layer1_kqvs(const float* __restrict__ x,
                            const float* __restrict__ W1,
                            const float* __restrict__ b1,
                            float* __restrict__ kqvs) {
  int i = blockIdx.x * blockDim.x + threadIdx.x;
  if (i >= N_NODES * KQVS) return;
  int n = i >> 8;
  int c = i & 255;
  kqvs[i] = x[n] * W1[c] + b1[c];
}

// ------------------------------------------ fused node GEMM via f32 WMMA
// kqvs[50000,256] = h[50000,64] @ Wcat[64,256] + bcat[256]
// Wcat column c maps to W[(c>>6)*4096 + k*64 + (c&63)] (W = W_stack[layer]).
// One wave per 16x16 output tile; K=64 swept with 16x V_WMMA_F32_16X16X4_F32.
__global__ void __launch_bounds__(256)
gemm_kqvs(const float* __restrict__ h,
          const float* __restrict__ W,   // [4,64,64] for this layer
          const float* __restrict__ b,   // [256]
          float* __restrict__ kqvs) {
  const int wave = threadIdx.x >> 5;
  const int lane = threadIdx.x & 31;
  const int tile = blockIdx.x * 8 + wave;               // wave-uniform
  if (tile >= (N_NODES / 16) * (KQVS / 16)) return;     // exact fit anyway

  const int mtile = tile >> 4;        // 0..3124
  const int ntile = tile & 15;        // 0..15
  const int node0 = mtile * 16;
  const int col0  = ntile * 16;

  const int half = lane >> 4;         // 0: K+{0,1}, 1: K+{2,3}
  const int l    = lane & 15;         // M index (A) / N index (B)
  const int koff = half * 2;

  const float* __restrict__ arow = h + (size_t)(node0 + l) * HID;
  const int col = col0 + l;
  const float* __restrict__ wcol = W + (size_t)(col >> 6) * 4096 + (col & 63);

  v8f acc = {};
#pragma unroll
  for (int kk = 0; kk < 16; ++kk) {
    const int k0 = kk * 4 + koff;
    v2f a, bb;
    a.x  = arow[k0];
    a.y  = arow[k0 + 1];
    bb.x = wcol[(size_t)k0 * 64];
    bb.y = wcol[(size_t)(k0 + 1) * 64];
    acc = __builtin_amdgcn_wmma_f32_16x16x4_f32(
        /*neg_a=*/false, a, /*neg_b=*/false, bb,
        /*c_mod=*/(short)0, acc, /*reuse_a=*/false, /*reuse_b=*/false);
  }

  // D layout: VGPR r -> row (r + half*8), lane&15 -> column
  const float bias = b[col];
#pragma unroll
  for (int r = 0; r < 8; ++r) {
    const int row = node0 + r + half * 8;
    kqvs[(size_t)row * KQVS + col] = acc[r] + bias;
  }
}

// ---------------------------------------------- edge gate + scatter-add
// 16 lanes per edge, float4 per lane. k/q/v tables are L2-resident (38 MB).
__global__ void edge_gate(const float* __restrict__ kqvs,
                          const int* __restrict__ ei,    // [2, N_EDGES]
                          float* __restrict__ aggr) {    // [N_NODES, 64]
  int i = blockIdx.x * blockDim.x + threadIdx.x;
  if (i >= N_EDGES * 16) return;
  const int e = i >> 4;
  const int f = (i & 15) << 2;
  const int s = ei[e];
  const int d = ei[N_EDGES + e];

  const float4 k = *(const float4*)(kqvs + (size_t)d * KQVS + f);
  const float4 q = *(const float4*)(kqvs + (size_t)s * KQVS + 64 + f);
  const float4 v = *(const float4*)(kqvs + (size_t)s * KQVS + 128 + f);

  float4 m;
  m.x = v.x / (1.0f + __expf(-(k.x + q.x)));
  m.y = v.y / (1.0f + __expf(-(k.y + q.y)));
  m.z = v.z / (1.0f + __expf(-(k.z + q.z)));
  m.w = v.w / (1.0f + __expf(-(k.w + q.w)));

  float* ap = aggr + (size_t)d * HID + f;
  unsafeAtomicAdd(ap + 0, m.x);
  unsafeAtomicAdd(ap + 1, m.y);
  unsafeAtomicAdd(ap + 2, m.z);
  unsafeAtomicAdd(ap + 3, m.w);
}

// ------------------------------------------------- h = skip + aggr (, relu)
__global__ void finalize_nodes(const float* __restrict__ kqvs,
                               const float* __restrict__ aggr,
                               float* __restrict__ h, int relu) {
  int i = blockIdx.x * blockDim.x + threadIdx.x;
  if (i >= N_NODES * HID) return;
  const int n = i >> 6;
  const int f = i & 63;
  float v = kqvs[(size_t)n * KQVS + 192 + f] + aggr[i];
  if (relu) v = fmaxf(v, 0.0f);
  h[i] = v;
}

// ----------------------------------------------------------- mean pool
__global__ void pool_accum(const float* __restrict__ h,
                           const int* __restrict__ batch,
                           float* __restrict__ sums,     // [G, 64]
                           float* __restrict__ counts) { // [G]
  int i = blockIdx.x * blockDim.x + threadIdx.x;
  if (i >= N_NODES * HID) return;
  const int n = i >> 6;
  const int f = i & 63;
  const int g = batch[n];
  unsafeAtomicAdd(&sums[(size_t)g * HID + f], h[i]);
  if (f == 0) unsafeAtomicAdd(&counts[g], 1.0f);
}

__global__ void pool_out(const float* __restrict__ sums,
                         const float* __restrict__ counts,
                         const float* __restrict__ Wlin,  // [64,10]
                         const float* __restrict__ blin,  // [10]
                         float* __restrict__ out) {       // [G,10]
  int i = blockIdx.x * blockDim.x + threadIdx.x;
  if (i >= N_GRAPHS * N_CLASSES) return;
  const int g = i / N_CLASSES;
  const int c = i % N_CLASSES;
  const float inv = 1.0f / fmaxf(counts[g], 1.0f);
  float acc = blin[c];
#pragma unroll
  for (int f = 0; f < HID; ++f)
    acc += sums[(size_t)g * HID + f] * inv * Wlin[f * N_CLASSES + c];
  out[i] = acc;
}

// ---------------------------------------------------------------- driver
extern "C" void kernel_launch(void* const* d_in, const int* in_sizes, int n_in,
                              void* d_out, int out_size, void* d_ws, size_t ws_size,
                              hipStream_t stream) {
  const float* x       = (const float*)d_in[0];
  const int*   ei      = (const int*)  d_in[1];
  const int*   batch   = (const int*)  d_in[2];
  const float* W1      = (const float*)d_in[3];  // [4,1,64] -> flat [256]
  const float* b1      = (const float*)d_in[4];  // [4,64]   -> flat [256]
  const float* W_stack = (const float*)d_in[5];  // [3,4,64,64]
  const float* b_stack = (const float*)d_in[6];  // [3,4,64]
  const float* Wlin    = (const float*)d_in[7];
  const float* blin    = (const float*)d_in[8];
  float* out = (float*)d_out;

  float* kqvs   = (float*)d_ws;                       // 50000*256
  float* aggr   = kqvs + (size_t)N_NODES * KQVS;      // 50000*64
  float* h      = aggr + (size_t)N_NODES * HID;       // 50000*64
  float* sums   = h    + (size_t)N_NODES * HID;       // 500*64
  float* counts = sums + (size_t)N_GRAPHS * HID;      // 500

  const int TB = 256;
  const int g_node_feat = (N_NODES * HID + TB - 1) / TB;    // 12500
  const int g_kqvs      = (N_NODES * KQVS + TB - 1) / TB;   // 50000
  const int g_edge      = (N_EDGES * 16 + TB - 1) / TB;     // 50000
  const int g_gemm      = (N_NODES / 16) * (KQVS / 16) / 8; // 6250

  // ---- layer 1 (1 -> 64)
  layer1_kqvs<<<g_kqvs, TB, 0, stream>>>(x, W1, b1, kqvs);
  zero_f32<<<g_node_feat, TB, 0, stream>>>(aggr, N_NODES * HID);
  edge_gate<<<g_edge, TB, 0, stream>>>(kqvs, ei, aggr);
  finalize_nodes<<<g_node_feat, TB, 0, stream>>>(kqvs, aggr, h, 1);

  // ---- layers 2..4 (64 -> 64), relu on first two
  for (int layer = 0; layer < 3; ++layer) {
    gemm_kqvs<<<g_gemm, TB, 0, stream>>>(h,
                                         W_stack + (size_t)layer * 4 * HID * HID,
                                         b_stack + (size_t)layer * 4 * HID,
                                         kqvs);
    zero_f32<<<g_node_feat, TB, 0, stream>>>(aggr, N_NODES * HID);
    edge_gate<<<g_edge, TB, 0, stream>>>(kqvs, ei, aggr);
    finalize_nodes<<<g_node_feat, TB, 0, stream>>>(kqvs, aggr, h, layer < 2 ? 1 : 0);
  }

  // ---- global mean pool + linear head
  zero_f32<<<(N_GRAPHS * HID + N_GRAPHS + TB - 1) / TB, TB, 0, stream>>>(
      sums, N_GRAPHS * HID + N_GRAPHS);
  pool_accum<<<g_node_feat, TB, 0, stream>>>(h, batch, sums, counts);
  pool_out<<<(N_GRAPHS * N_CLASSES + TB - 1) / TB, TB, 0, stream>>>(
      sums, counts, Wlin, blin, out);
}